// VolumetricSampler_19868518711413
// MI455X (gfx1250) — compile-verified
//
#include <hip/hip_runtime.h>
#include <hip/hip_bf16.h>
#include <stdint.h>

#define N_RAYS   65536
#define S        256
#define GRID_RES 128
#define ALPHA    0.01f

// STEP = 2*sqrt(3)/256
__device__ __constant__ const float kStep = 2.0f * 1.7320508075688772f / 256.0f;

typedef float __attribute__((ext_vector_type(4))) float4v;

__global__ __launch_bounds__(256) void volsampler_kernel(
    const float* __restrict__ rays_o,
    const float* __restrict__ rays_d,
    const float* __restrict__ occ,
    const float* __restrict__ aabb,
    float*       __restrict__ out)
{
    const int lane = threadIdx.x & 31;   // wave32
    const int wave = threadIdx.x >> 5;   // 8 waves / block
    const int ray  = (int)blockIdx.x * 8 + wave;

    // ---- Stage this block's 8 rays (24 floats o + 24 floats d) into LDS via
    // ---- the CDNA5 async-to-LDS engine (ASYNCcnt). 12 lanes x b128.
    // ---- 16B alignment required for the B128 LDS writes.
    __shared__ __align__(16) float lds_rays[64];
    if (wave == 0 && lane < 12) {
        unsigned lds_off = ((unsigned)(uintptr_t)&lds_rays[0]) + (unsigned)lane * 16u;
        const float* gsrc = (lane < 6)
            ? (rays_o + (size_t)blockIdx.x * 24 + (size_t)lane * 4)
            : (rays_d + (size_t)blockIdx.x * 24 + (size_t)(lane - 6) * 4);
        asm volatile("global_load_async_to_lds_b128 %0, %1, off"
                     :: "v"(lds_off), "v"(gsrc)
                     : "memory");
    }
    asm volatile("s_wait_asynccnt 0" ::: "memory");
    __syncthreads();

    // Per-wave ray data from LDS
    const float ox = lds_rays[wave * 3 + 0];
    const float oy = lds_rays[wave * 3 + 1];
    const float oz = lds_rays[wave * 3 + 2];
    const float dx = lds_rays[24 + wave * 3 + 0];
    const float dy = lds_rays[24 + wave * 3 + 1];
    const float dz = lds_rays[24 + wave * 3 + 2];

    // AABB (uniform -> scalar loads)
    const float mnx = aabb[0], mny = aabb[1], mnz = aabb[2];
    const float mxx = aabb[3], mxy = aabb[4], mxz = aabb[5];

    // Ray/AABB intersection
    const float ivx = 1.0f / dx, ivy = 1.0f / dy, ivz = 1.0f / dz;
    const float t0x = (mnx - ox) * ivx, t1x = (mxx - ox) * ivx;
    const float t0y = (mny - oy) * ivy, t1y = (mxy - oy) * ivy;
    const float t0z = (mnz - oz) * ivz, t1z = (mxz - oz) * ivz;
    const float t_enter = fmaxf(fmaxf(fminf(t0x, t1x), fminf(t0y, t1y)), fminf(t0z, t1z));
    const float t_exit  = fminf(fminf(fmaxf(t0x, t1x), fmaxf(t0y, t1y)), fmaxf(t0z, t1z));
    const float t_min   = fmaxf(t_enter, 0.0f);
    const bool  hit     = t_exit > t_min;
    // Samples with end <= t_lim are in range; fold `hit` in so the inner
    // predicate is a single compare.
    const float t_lim   = hit ? t_exit : -1e30f;

    // Grid mapping scale
    const float sclx = (float)GRID_RES / (mxx - mnx);
    const float scly = (float)GRID_RES / (mxy - mny);
    const float sclz = (float)GRID_RES / (mxz - mnz);

    // Output sections (reference tuple order), flat float32
    const size_t NS = (size_t)N_RAYS * S;
    float* o_orig = out;
    float* o_dirs = out + NS * 3;
    float* o_sta  = out + NS * 6;
    float* o_end  = o_sta + NS;
    float* o_rid  = o_end + NS;
    float* o_msk  = o_rid + NS;

    const float STEP = kStep;

    // ---- origins / dirs: period-3 repeating float stream, 6 NT b128 stores each
    const float4v P0 = {ox, oy, oz, ox}, P1 = {oy, oz, ox, oy}, P2 = {oz, ox, oy, oz};
    const float4v Q0 = {dx, dy, dz, dx}, Q1 = {dy, dz, dx, dy}, Q2 = {dz, dx, dy, dz};
    float4v* op = (float4v*)(o_orig + (size_t)ray * S * 3);
    float4v* dp = (float4v*)(o_dirs + (size_t)ray * S * 3);
#pragma unroll
    for (int t = 0; t < 6; ++t) {
        const int i = 32 * t + lane;          // float4 chunk index within ray (0..191)
        const int p = i % 3;                  // phase of repeating pattern
        const float4v Pv = (p == 0) ? P0 : ((p == 1) ? P1 : P2);
        const float4v Qv = (p == 0) ? Q0 : ((p == 1) ? Q1 : Q2);
        __builtin_nontemporal_store(Pv, op + i);
        __builtin_nontemporal_store(Qv, dp + i);
    }

    // ---- starts / ends / ray_indices / mask: lane owns chunks {lane, lane+32}
    const size_t rbase = (size_t)ray * S;
    float4v* stp = (float4v*)(o_sta + rbase);
    float4v* enp = (float4v*)(o_end + rbase);
    float4v* rip = (float4v*)(o_rid + rbase);
    float4v* mkp = (float4v*)(o_msk + rbase);
    const float frid = (float)ray;
    const float4v ridv = {frid, frid, frid, frid};

#pragma unroll
    for (int h = 0; h < 2; ++h) {
        const int   c  = lane + h * 32;       // float4 chunk of samples 4c..4c+3
        const float s0 = (float)(4 * c);
        float4v st, en, mk;
#pragma unroll
        for (int j = 0; j < 4; ++j) {
            const float stj = fmaf(s0 + (float)j, STEP, t_min);
            const float enj = stj + STEP;
            st[j] = stj;
            en[j] = enj;
            float m = 0.0f;
            if (enj <= t_lim) {               // in_range && hit (EXEC-masked gather)
                const float mid = stj + 0.5f * STEP;
                const float px = fmaf(dx, mid, ox);
                const float py = fmaf(dy, mid, oy);
                const float pz = fmaf(dz, mid, oz);
                int ix = (int)floorf((px - mnx) * sclx);
                int iy = (int)floorf((py - mny) * scly);
                int iz = (int)floorf((pz - mnz) * sclz);
                ix = min(max(ix, 0), GRID_RES - 1);
                iy = min(max(iy, 0), GRID_RES - 1);
                iz = min(max(iz, 0), GRID_RES - 1);
                const float ov = occ[((size_t)ix * GRID_RES + (size_t)iy) * GRID_RES + (size_t)iz];
                m = (ov > ALPHA) ? 1.0f : 0.0f;
            }
            mk[j] = m;
        }
        __builtin_nontemporal_store(st,   stp + c);
        __builtin_nontemporal_store(en,   enp + c);
        __builtin_nontemporal_store(ridv, rip + c);
        __builtin_nontemporal_store(mk,   mkp + c);
    }
}

extern "C" void kernel_launch(void* const* d_in, const int* in_sizes, int n_in,
                              void* d_out, int out_size, void* d_ws, size_t ws_size,
                              hipStream_t stream) {
    const float* rays_o = (const float*)d_in[0];
    const float* rays_d = (const float*)d_in[1];
    const float* occ    = (const float*)d_in[2];
    const float* aabb   = (const float*)d_in[3];
    float* out = (float*)d_out;
    (void)in_sizes; (void)n_in; (void)out_size; (void)d_ws; (void)ws_size;

    // 8 rays per block (1 ray per wave32), 256 threads/block
    dim3 grid(N_RAYS / 8), block(256);
    volsampler_kernel<<<grid, block, 0, stream>>>(rays_o, rays_d, occ, aabb, out);
}